// LatentHandler_87591563034799
// MI455X (gfx1250) — compile-verified
//
#include <hip/hip_runtime.h>

namespace {
constexpr int BN     = 64;
constexpr int NN     = 16384;
constexpr int DWHAT  = 64;
constexpr int MAXOBJ = 512;
constexpr int NEGADD = 1638;            // int(0.1 * N)
constexpr int KK     = MAXOBJ + NEGADD; // 2150
constexpr int NT     = 1024;            // 32 wave32s per select block
constexpr int EPT    = NN / NT;         // 16 contiguous elements per thread
}

typedef __attribute__((address_space(3))) float lds_f32;

__device__ __forceinline__ unsigned waveInclScan(unsigned x) {
  const unsigned lane = threadIdx.x & 31u;
#pragma unroll
  for (int off = 1; off < 32; off <<= 1) {
    unsigned y = __shfl_up(x, (unsigned)off, 32);
    if (lane >= (unsigned)off) x += y;
  }
  return x;
}

// Exclusive block scan over one value per thread; also returns block total.
__device__ __forceinline__ unsigned blockExclScan(unsigned v, unsigned* wsum, unsigned* total) {
  __syncthreads();  // protect wsum reuse across calls
  const unsigned lane = threadIdx.x & 31u;
  const unsigned wid  = threadIdx.x >> 5;
  unsigned inc = waveInclScan(v);
  if (lane == 31u) wsum[wid] = inc;
  __syncthreads();
  if (wid == 0) {
    unsigned w    = wsum[lane];
    unsigned winc = waveInclScan(w);
    wsum[lane] = winc - w;                 // exclusive wave offsets
    if (lane == 31u) wsum[32] = winc;      // block total
  }
  __syncthreads();
  *total = wsum[32];
  return wsum[wid] + inc - v;
}

__device__ __forceinline__ void gather_row(int b, int s, int j, float mod, int l,
    const float* __restrict__ zw, const float* __restrict__ zp,
    const float* __restrict__ zwhat, const float* __restrict__ zd,
    float* __restrict__ out) {
  const size_t rowN = (size_t)b * NN + (size_t)j;
  const size_t rowK = (size_t)b * KK + (size_t)s;
  float* o0 = out;                                   // (B,K,4)  z_where
  float* o1 = out + (size_t)BN * KK * 4;             // (B,K,1)  modified
  float* o2 = o1 + (size_t)BN * KK;                  // (B,K,64) masked what
  float* o3 = o2 + (size_t)BN * KK * DWHAT;          // (B,K,1)  masked depth
  __builtin_prefetch(zwhat + rowN * DWHAT + l, 0, 0);  // global_prefetch_b8
  const float p   = zp[rowN];
  const bool mask = p > 0.001f;                      // EPS
  const float w   = zwhat[rowN * DWHAT + l];
  o2[rowK * DWHAT + l] = mask ? w : 0.0f;
  if (l < 4)  o0[rowK * 4 + l] = zw[rowN * 4 + l];
  if (l == 0) { o1[rowK] = mod; o3[rowK] = mask ? zd[rowN] : 0.0f; }
}

__global__ __launch_bounds__(NT, 1) void select_kernel(
    const float* __restrict__ zw, const float* __restrict__ zp,
    const float* __restrict__ zwhat, const float* __restrict__ zd,
    float* __restrict__ out, int* __restrict__ wsIdx, float* __restrict__ wsMod,
    int useWs)
{
  __shared__ float          sp[NN];        // 64KB: p for this batch
  __shared__ unsigned       s_hist[256];
  __shared__ unsigned       s_wsum[33];
  __shared__ unsigned       s_sel[4];      // 0:bin 1:want_next 2:modeAll 3:total_present
  __shared__ unsigned short s_idx[KK];
  __shared__ float          s_mod[KK];

  const int tid = (int)threadIdx.x;
  const int b   = (int)blockIdx.x;
  const float* src = zp + (size_t)b * NN;

  // ---- CDNA5 async global->LDS copy of z_present[b,:,0] (16384 f32) ----
#pragma unroll
  for (int k = 0; k < (NN / 4) / NT; ++k) {          // 4 x b128 per thread
    const int o = tid + k * NT;                      // 0..4095, 16B each
    unsigned long long ga = (unsigned long long)(const void*)(src + (size_t)o * 4);
    unsigned la = (unsigned)(size_t)(lds_f32*)(sp + o * 4);
    asm volatile("global_load_async_to_lds_b128 %0, %1, off" :: "v"(la), "v"(ga) : "memory");
  }
  asm volatile("s_wait_asynccnt 0" ::: "memory");
  if (tid == 0) s_sel[2] = 0u;
  __syncthreads();

  // ---- exact radix-select of the 512th-largest present p (float bits) ----
  unsigned prefix = 0u, prefmask = 0u;
  int want = MAXOBJ;
  bool modeAll = false;

#pragma unroll 1
  for (int pass = 0; pass < 4; ++pass) {
    const int shift = 24 - 8 * pass;
    for (int i = tid; i < 256; i += NT) s_hist[i] = 0u;
    __syncthreads();
    if (!modeAll) {
#pragma unroll
      for (int k = 0; k < EPT; ++k) {
        const float p = sp[tid * EPT + k];
        if (p > 0.5f) {
          const unsigned key = __float_as_uint(p);
          if ((key & prefmask) == prefix)
            atomicAdd(&s_hist[(key >> shift) & 255u], 1u);
        }
      }
    }
    __syncthreads();
    if (tid == 0 && s_sel[2] == 0u) {
      unsigned cum = 0u, above = 0u; int bin = 0; bool found = false;
      for (int bq = 255; bq >= 0; --bq) {
        const unsigned h = s_hist[bq];
        if (!found && cum + h >= (unsigned)want) { bin = bq; above = cum; found = true; }
        cum += h;
      }
      if (pass == 0) {
        s_sel[3] = cum;                           // total present
        if (cum <= (unsigned)MAXOBJ) s_sel[2] = 1u;  // keep all present
      }
      s_sel[0] = (unsigned)bin;
      s_sel[1] = (unsigned)(want - (int)above);
    }
    __syncthreads();
    modeAll = (s_sel[2] != 0u);
    if (!modeAll) {
      prefix   |= s_sel[0] << shift;
      prefmask |= 255u << shift;
      want      = (int)s_sel[1];
    }
  }
  const unsigned mkey  = prefix;   // threshold key (exact value bits)
  const int     wantEq = want;     // #elements equal to threshold to keep (lowest indices)

  const int basei = tid * EPT;

  // ---- eq-rank scan for exact tie handling ----
  unsigned eqc = 0u;
  if (!modeAll) {
#pragma unroll
    for (int k = 0; k < EPT; ++k) {
      const float p = sp[basei + k];
      if (p > 0.5f && __float_as_uint(p) == mkey) eqc++;
    }
  }
  unsigned totEq;
  const unsigned eqbase = blockExclScan(eqc, s_wsum, &totEq);
  (void)totEq;

  // ---- keep flags ----
  unsigned keepbits = 0u, keepc = 0u;
  {
    unsigned eqseen = 0u;
#pragma unroll
    for (int k = 0; k < EPT; ++k) {
      const float p   = sp[basei + k];
      const bool pres = p > 0.5f;
      bool kp;
      if (modeAll) kp = pres;
      else {
        const unsigned key = __float_as_uint(p);
        const bool eq = pres && (key == mkey);
        kp = pres && (key > mkey || (eq && (int)(eqbase + eqseen) < wantEq));
        eqseen += eq ? 1u : 0u;
      }
      if (kp) { keepbits |= 1u << k; keepc++; }
    }
  }
  unsigned nkeep;
  const unsigned kbase = blockExclScan(keepc, s_wsum, &nkeep);
  const int negcut = ((int)NN - (int)nkeep) - NEGADD;  // last NEGADD zeros -> -1

  // ---- element classes {0:pad, 1:keep, 2:negative}; count nonzero ----
  unsigned flag2 = 0u, nzc = 0u;
  {
    unsigned kseen = 0u;
#pragma unroll
    for (int k = 0; k < EPT; ++k) {
      const bool kp = (keepbits >> k) & 1u;
      unsigned f = 1u;
      if (!kp) {
        const int zrank = (basei + k) - (int)(kbase + kseen);  // exclusive zero-rank
        f = (zrank >= negcut) ? 2u : 0u;
      } else kseen++;
      if (f != 0u) nzc++;
      flag2 |= f << (2 * k);
    }
  }
  unsigned totNz;
  const unsigned nzbase = blockExclScan(nzc, s_wsum, &totNz);

  // ---- emit compacted idx/modified lists (stable order) ----
  {
    unsigned nzseen = 0u;
#pragma unroll
    for (int k = 0; k < EPT; ++k) {
      const unsigned f = (flag2 >> (2 * k)) & 3u;
      const int i = basei + k;
      if (f != 0u) {
        const unsigned pos = nzbase + nzseen;
        if (pos < (unsigned)KK) { s_idx[pos] = (unsigned short)i; s_mod[pos] = (f == 1u) ? 1.0f : -1.0f; }
        nzseen++;
      } else {
        const unsigned slot = totNz + ((unsigned)i - (nzbase + nzseen)); // pad tail
        if (slot < (unsigned)KK) { s_idx[slot] = (unsigned short)i; s_mod[slot] = 0.0f; }
      }
    }
  }
  __syncthreads();

  if (useWs) {
    for (int q = tid; q < KK; q += NT) {
      wsIdx[(size_t)b * KK + q] = (int)s_idx[q];
      wsMod[(size_t)b * KK + q] = s_mod[q];
    }
  } else {
    const int l = tid & 63;
    for (int s0 = tid >> 6; s0 < KK; s0 += NT / 64) {
      gather_row(b, s0, (int)s_idx[s0], s_mod[s0], l, zw, zp, zwhat, zd, out);
    }
  }
}

__global__ __launch_bounds__(256, 1) void gather_kernel(
    const float* __restrict__ zw, const float* __restrict__ zp,
    const float* __restrict__ zwhat, const float* __restrict__ zd,
    const int* __restrict__ wsIdx, const float* __restrict__ wsMod,
    float* __restrict__ out)
{
  const int r = (int)blockIdx.x * 4 + ((int)threadIdx.x >> 6);
  if (r >= BN * KK) return;
  const int l = (int)threadIdx.x & 63;
  const int b = r / KK;
  const int s = r - b * KK;
  gather_row(b, s, wsIdx[r], wsMod[r], l, zw, zp, zwhat, zd, out);
}

extern "C" void kernel_launch(void* const* d_in, const int* in_sizes, int n_in,
                              void* d_out, int out_size, void* d_ws, size_t ws_size,
                              hipStream_t stream) {
  (void)in_sizes; (void)n_in; (void)out_size;
  const float* zw    = (const float*)d_in[0];  // z_where     (B,N,4)
  const float* zp    = (const float*)d_in[1];  // z_present   (B,N,1)
  const float* zwhat = (const float*)d_in[2];  // z_what_loc  (B,N,64)
  const float* zd    = (const float*)d_in[4];  // z_depth_loc (B,N,1)
  float* out = (float*)d_out;

  const size_t need = (size_t)BN * KK * (sizeof(int) + sizeof(float));
  if (d_ws != nullptr && ws_size >= need) {
    int*   wsIdx = (int*)d_ws;
    float* wsMod = (float*)((char*)d_ws + (size_t)BN * KK * sizeof(int));
    select_kernel<<<BN, NT, 0, stream>>>(zw, zp, zwhat, zd, out, wsIdx, wsMod, 1);
    gather_kernel<<<(BN * KK + 3) / 4, 256, 0, stream>>>(zw, zp, zwhat, zd, wsIdx, wsMod, out);
  } else {
    select_kernel<<<BN, NT, 0, stream>>>(zw, zp, zwhat, zd, out, nullptr, nullptr, 0);
  }
}